// BioTUCHLoss_66039417143831
// MI455X (gfx1250) — compile-verified
//
#include <hip/hip_runtime.h>
#include <hip/hip_bf16.h>

typedef __attribute__((ext_vector_type(2))) float v2f;
typedef __attribute__((ext_vector_type(8))) float v8f;

#define HBINS     4096        // d2 float bits >> 19 : sign+exp+4 mantissa bits
#define HSHIFT    19
#define CAND_CAP  131072u
#define SORT_N    8192        // 8192 * 8B = 64KB LDS for the one-block sort

// -------- Pass 0: pack (x,y,z,||v||^2) float4s + zero histogram/counter ---
__global__ void init_kernel(const float* __restrict__ hand, const float* __restrict__ body,
                            float4* __restrict__ hand4, float4* __restrict__ body4,
                            unsigned* __restrict__ hist, unsigned* __restrict__ cnt,
                            int Nh, int Nb) {
    int i = blockIdx.x * blockDim.x + threadIdx.x;
    if (i < Nh) {
        float x = hand[i * 3 + 0], y = hand[i * 3 + 1], z = hand[i * 3 + 2];
        hand4[i] = make_float4(x, y, z, x * x + y * y + z * z);
    }
    int j = i - Nh;
    if (j >= 0 && j < Nb) {
        float x = body[j * 3 + 0], y = body[j * 3 + 1], z = body[j * 3 + 2];
        body4[j] = make_float4(x, y, z, x * x + y * y + z * z);
    }
    int h = i - Nh - Nb;
    if (h >= 0 && h < HBINS) hist[h] = 0u;
    if (i == Nh + Nb + HBINS) *cnt = 0u;
}

// -------- Passes 1 & 3: WMMA emits complete d2 tiles ----------------------
// K=4 slots: K=0..2 carry -2*hand coords; K=3 carries A=1.0, B=||b||^2 so the
// matrix unit itself broadcasts the body norm.  C = ||h_m||^2 (loop-invariant
// registers).  D = (-2*h)·b + ||b||^2 + ||h||^2 = d2, no per-tile epilogue
// arithmetic except the bin index.  sqrt is monotonic so d2-bit ordering ==
// distance ordering; sqrt is never computed.
template <int MODE>   // 0: histogram, 1: threshold compaction
__global__ __launch_bounds__(256)
void dist_tiles(const float4* __restrict__ hand4, const float4* __restrict__ body4,
                unsigned* __restrict__ hist, unsigned* __restrict__ cnt,
                const unsigned* __restrict__ thresh, uint2* __restrict__ cand,
                unsigned cap, int Nh, int Nb) {
    __shared__ unsigned lhist[MODE == 0 ? HBINS : 1];
    if (MODE == 0) {
        for (int i = threadIdx.x; i < HBINS; i += 256) lhist[i] = 0u;
        __syncthreads();
    }
    const int lane = threadIdx.x & 31;
    const int wave = threadIdx.x >> 5;
    const int wgv  = blockIdx.x * 8 + wave;          // global wave id
    const int strips  = Nb >> 8;                     // 256 body cols per wave
    const int rowTile = wgv / strips;
    const int strip   = wgv % strips;
    const int m0   = rowTile << 4;
    const int col  = lane & 15;
    const bool hi  = lane >= 16;                     // lanes 0-15: K=0,1  16-31: K=2,3

    // A fragment: K=0..2 = -2*hand coords, K=3 = 1.0 (norm-broadcast lane)
    const float4 ha = hand4[m0 + col];
    v2f a;
    a.x = -2.0f * (hi ? ha.z : ha.x);
    a.y = hi ? 1.0f : (-2.0f * ha.y);

    // C operand: ||h||^2 for this lane's 8 rows, kept in registers all loop
    const int mbase = m0 + (hi ? 8 : 0);
    v8f hqv;
#pragma unroll
    for (int v = 0; v < 8; ++v) hqv[v] = hand4[mbase + v].w;

    unsigned tbits = 0xFFFFFFFFu;
    if (MODE == 1) tbits = *thresh;

    for (int t = 0; t < 16; ++t) {
        const int j0 = (strip << 8) + (t << 4);
        const int bc = j0 + col;
        // One b128 per lane per tile; K=3 row of B carries ||b||^2
        const float4 bv = body4[bc];
        v2f b;
        b.x = hi ? bv.z : bv.x;
        b.y = hi ? bv.w : bv.y;

        v8f acc = __builtin_amdgcn_wmma_f32_16x16x4_f32(
            /*neg_a=*/false, a, /*neg_b=*/false, b,
            /*c_mod=*/(short)0, hqv, /*reuse_a=*/false, /*reuse_b=*/false);

#pragma unroll
        for (int v = 0; v < 8; ++v) {
            unsigned bits = __float_as_uint(acc[v]);
            bits &= ~(unsigned)((int)bits >> 31);     // negative d2 -> +0.0 bits
            if (MODE == 0) {
                atomicAdd(&lhist[bits >> HSHIFT], 1u);
            } else {
                if (bits <= tbits) {
                    unsigned pos = atomicAdd(cnt, 1u);
                    if (pos < cap) {
                        unsigned flat = (unsigned)(mbase + v) * (unsigned)Nb + (unsigned)bc;
                        cand[pos] = make_uint2(bits, flat);
                    }
                }
            }
        }
    }
    if (MODE == 0) {
        __syncthreads();
        for (int i = threadIdx.x; i < HBINS; i += 256) {
            unsigned c = lhist[i];
            if (c) atomicAdd(&hist[i], c);
        }
    }
}

// -------- Pass 2: find bin containing the k-th smallest d2 ----------------
__global__ void thresh_kernel(const unsigned* __restrict__ hist,
                              unsigned* __restrict__ thresh,
                              const int* __restrict__ topk) {
    unsigned need = (unsigned)*topk;
    unsigned sum = 0u, tb = 0xFFFFFFFFu;
    for (int b = 0; b < HBINS; ++b) {
        sum += hist[b];
        if (sum >= need) { tb = (((unsigned)(b + 1)) << HSHIFT) - 1u; break; }
    }
    *thresh = tb;
}

// -------- Pass 4: one-block bitonic sort + weighted-L1 gather -------------
__global__ __launch_bounds__(1024)
void select_kernel(const float* __restrict__ hand, const float* __restrict__ body,
                   const float* __restrict__ w, const uint2* __restrict__ cand,
                   const unsigned* __restrict__ cnt, float* __restrict__ out,
                   unsigned cap, int K, int Nb) {
    __shared__ unsigned long long keys[SORT_N];
    unsigned n = *cnt;
    if (n > cap) n = cap;
    if (n > SORT_N) n = SORT_N;
    for (int i = threadIdx.x; i < SORT_N; i += 1024) {
        if (i < (int)n) {
            uint2 c = cand[i];
            keys[i] = ((unsigned long long)c.x << 32) | (unsigned long long)c.y;
        } else {
            keys[i] = ~0ull;
        }
    }
    __syncthreads();
    for (int k = 2; k <= SORT_N; k <<= 1) {
        for (int j = k >> 1; j > 0; j >>= 1) {
            for (int i = threadIdx.x; i < SORT_N; i += 1024) {
                int ixj = i ^ j;
                if (ixj > i) {
                    bool up = ((i & k) == 0);
                    unsigned long long A = keys[i], B = keys[ixj];
                    if ((up && A > B) || (!up && A < B)) { keys[i] = B; keys[ixj] = A; }
                }
            }
            __syncthreads();
        }
    }
    const float w0 = w[0], w1 = w[1], w2 = w[2];
    for (int k = threadIdx.x; k < K; k += 1024) {
        unsigned long long key = keys[k];
        float val = 0.0f;
        if (key != ~0ull) {
            unsigned flat = (unsigned)(key & 0xFFFFFFFFu);
            unsigned h = flat / (unsigned)Nb;
            unsigned b = flat % (unsigned)Nb;
            float dx = fabsf(hand[h * 3 + 0] - body[b * 3 + 0]);
            float dy = fabsf(hand[h * 3 + 1] - body[b * 3 + 1]);
            float dz = fabsf(hand[h * 3 + 2] - body[b * 3 + 2]);
            val = w0 * dx + w1 * dy + w2 * dz;
        }
        out[k] = val;
    }
}

extern "C" void kernel_launch(void* const* d_in, const int* in_sizes, int n_in,
                              void* d_out, int out_size, void* d_ws, size_t ws_size,
                              hipStream_t stream) {
    const float* hand = (const float*)d_in[0];
    const float* body = (const float*)d_in[1];
    const float* wsel = (const float*)d_in[2];
    const int*   topk = (const int*)d_in[3];
    const int Nh = in_sizes[0] / 3;   // 4096
    const int Nb = in_sizes[1] / 3;   // 32768

    // Workspace layout (d_ws assumed >= 16B aligned)
    float4*   hand4  = (float4*)d_ws;
    float4*   body4  = hand4 + Nh;
    unsigned* hist   = (unsigned*)(body4 + Nb);
    unsigned* cnt    = hist + HBINS;
    unsigned* thresh = cnt + 1;
    size_t candOff   = ((size_t)(Nh + Nb) * 16 + (size_t)(HBINS + 2) * 4 + 7) & ~(size_t)7;
    uint2*    cand   = (uint2*)((char*)d_ws + candOff);
    unsigned  cap    = CAND_CAP;
    if (ws_size > candOff) {
        size_t avail = (ws_size - candOff) / sizeof(uint2);
        if (avail < (size_t)cap) cap = (unsigned)avail;
    } else {
        cap = 0;
    }
    float* out = (float*)d_out;

    const int initN  = Nh + Nb + HBINS + 1;
    init_kernel<<<(initN + 255) / 256, 256, 0, stream>>>(hand, body, hand4, body4, hist, cnt, Nh, Nb);

    const int waves  = (Nh / 16) * (Nb / 256);  // one wave per 16x256 strip
    const int blocks = waves / 8;               // 8 waves (256 threads) per block
    dist_tiles<0><<<blocks, 256, 0, stream>>>(hand4, body4, hist, cnt, thresh, cand, cap, Nh, Nb);
    thresh_kernel<<<1, 1, 0, stream>>>(hist, thresh, topk);
    dist_tiles<1><<<blocks, 256, 0, stream>>>(hand4, body4, hist, cnt, thresh, cand, cap, Nh, Nb);
    select_kernel<<<1, 1024, 0, stream>>>(hand, body, wsel, cand, cnt, out, cap, out_size, Nb);
}